// BlockSelector_79087527788599
// MI455X (gfx1250) — compile-verified
//
#include <hip/hip_runtime.h>

// BlockSelector: pure int32 index arithmetic, streaming / bandwidth bound.
// Inputs (setup_inputs order): 0:q 1:k 2:v 3:topk_indices 4:query_block_indices
//                              5:block_size(scalar) 6:seq_len(scalar)
// q/k/v are never read (reference uses them only for shapes).
// d_out = float32 concat: self_start | self_end | moba_start | moba_end | moba_valid
//
// One lane per (b,h,s): 2 self stores + K(=3) contiguous moba loads/stores.
// All streams touched exactly once -> non-temporal hints; prefetch topk ahead.

__global__ __launch_bounds__(256) void
blocksel_kernel(const int* __restrict__ topk,   // [B*H*S*K]
                const int* __restrict__ qbi,    // [B*H*S] current block index
                const int* __restrict__ bs_p,   // [1] block_size
                const int* __restrict__ sl_p,   // [1] seq_len
                float* __restrict__ out,
                int selfN, int K) {
  const int t = blockIdx.x * blockDim.x + threadIdx.x;
  if (t >= selfN) return;

  // Uniform scalar loads (s_load, KMcnt)
  const int bs = bs_p[0];
  const int sl = sl_p[0];

  // Prefetch the topk stream one block-stride ahead (global_prefetch_b8).
  {
    const int pf = (t + 256) * K;
    if (pf < selfN * K) __builtin_prefetch(&topk[pf], 0, 0);
  }

  const int cb = qbi[t];                    // current block index
  const int i  = t % sl;                    // sequence position within [B,H,·]
  const int cur_end = min((cb + 1) * bs, sl);

  float* __restrict__ self_start = out;
  float* __restrict__ self_end   = out + selfN;
  float* __restrict__ moba_start = out + 2 * selfN;
  float* __restrict__ moba_end   = moba_start + selfN * K;
  float* __restrict__ moba_valid = moba_end + selfN * K;

  __builtin_nontemporal_store((float)(cb * bs), &self_start[t]);
  __builtin_nontemporal_store((float)min(i + 1, cur_end), &self_end[t]);

  const int base = t * K;
  for (int kk = 0; kk < K; ++kk) {
    const int idx = __builtin_nontemporal_load(&topk[base + kk]);
    const bool valid = idx < cb;            // strictly-past blocks only
    const int ms = valid ? idx * bs : 0;
    const int me = valid ? min((idx + 1) * bs, sl) : 0;
    __builtin_nontemporal_store((float)ms, &moba_start[base + kk]);
    __builtin_nontemporal_store((float)me, &moba_end[base + kk]);
    __builtin_nontemporal_store(valid ? 1.0f : 0.0f, &moba_valid[base + kk]);
  }
}

extern "C" void kernel_launch(void* const* d_in, const int* in_sizes, int n_in,
                              void* d_out, int out_size, void* d_ws, size_t ws_size,
                              hipStream_t stream) {
  (void)n_in; (void)out_size; (void)d_ws; (void)ws_size;
  const int* topk = (const int*)d_in[3];
  const int* qbi  = (const int*)d_in[4];
  const int* bs_p = (const int*)d_in[5];
  const int* sl_p = (const int*)d_in[6];
  float* out = (float*)d_out;

  const int selfN = in_sizes[4];                 // B*H*S
  const int K     = in_sizes[3] / in_sizes[4];   // top-k

  const int threads = 256;                       // 8 wave32s / block
  const int blocks  = (selfN + threads - 1) / threads;
  blocksel_kernel<<<blocks, threads, 0, stream>>>(topk, qbi, bs_p, sl_p,
                                                  out, selfN, K);
}